// ElboGenerativeModelTop_39977555591236
// MI455X (gfx1250) — compile-verified
//
#include <hip/hip_runtime.h>
#include <math.h>

// Problem constants (fixed by the harness' setup_inputs()):
//   T = 256, BS = 128, N = 2048
#define BS_    128
#define N_     2048
#define BLOCK_ 128                 // 4 wave32s per batch row
#define NPT_   (N_ / BLOCK_)       // 16 features per thread, register resident
#define NWAVES (BLOCK_ / 32)
#define PF_AHEAD 8                 // L2 prefetch distance (covers HBM latency)

// CDNA5 split workgroup barrier: signal (after LDS ops drain), later wait.
__device__ __forceinline__ void wg_barrier_signal() {
    asm volatile("s_wait_dscnt 0x0\n\t"
                 "s_barrier_signal -1" ::: "memory");
}
__device__ __forceinline__ void wg_barrier_wait() {
    asm volatile("s_barrier_wait -1" ::: "memory");
}

__global__ __launch_bounds__(BLOCK_, 1)
void elbo_scan_kernel(const float* __restrict__ noises,   // [T, BS]
                      const float* __restrict__ ys,       // [T, BS]
                      const float* __restrict__ qs,       // [T, BS]
                      const float* __restrict__ z_biases, // [N]
                      const float* __restrict__ w_in_g,   // [N]
                      const float* __restrict__ w_inq_g,  // [N]
                      const float* __restrict__ p_llr,
                      const float* __restrict__ p_llrd,
                      const float* __restrict__ p_sigb,
                      const float* __restrict__ p_os,
                      const float* __restrict__ p_ufs,
                      const float* __restrict__ p_spwd,
                      const float* __restrict__ p_qsc,
                      const float* __restrict__ p_tq,
                      const float* __restrict__ p_ty,
                      const float* __restrict__ p_te,
                      float* __restrict__ out,            // [T, BS]
                      int T)
{
    const int b    = blockIdx.x;          // batch row, 0..127
    const int tid  = threadIdx.x;         // 0..127
    const int wave = tid >> 5;
    const int lane = tid & 31;

    // Double-buffered cross-wave partials -> one split barrier per timestep.
    __shared__ float2 partials[2][NWAVES];

    // ---- uniform scalar parameters (libm once, full accuracy) ----------
    const float lr0  = expf(p_llr[0]);
    const float lrd  = expf(p_llrd[0]);
    const float sigb = p_sigb[0];
    const float os   = p_os[0];
    const float ufs  = p_ufs[0];
    const float spwd = log1pf(expf(p_spwd[0]));           // softplus
    const float qsc  = p_qsc[0];
    const float tq   = 1.0f + log1pf(expf(p_tq[0]));
    const float ty   = 1.0f + log1pf(expf(p_ty[0]));
    const float te   = 1.0f + log1pf(expf(p_te[0]));
    const float aq = 1.0f - 1.0f / tq, bq = 1.0f / tq;
    const float ay = 1.0f - 1.0f / ty, by = 1.0f / ty;
    const float ae = 1.0f - 1.0f / te, be = 1.0f / te;

    // ---- register-resident per-feature state ---------------------------
    float bias[NPT_], win[NPT_], winq[NPT_], w[NPT_], h[NPT_];
#pragma unroll
    for (int k = 0; k < NPT_; ++k) {
        const int j = k * BLOCK_ + tid;   // coalesced constant loads
        bias[k] = sigb * z_biases[j];
        win[k]  = w_in_g[j];
        winq[k] = qsc * w_inq_g[j];       // fold q_scale into w_inq
        w[k]    = 0.0f;                   // w_out init
    }

    // ---- per-row scalar carry (replicated across all threads) ----------
    float u = 0.0f, e = 0.0f, lrm = 1.0f, qlp = 0.0f, ylp = 0.0f, elp = 0.0f;

    // Preload t = 0 inputs.
    float y  = ys    [b];
    float nx = noises[b];
    float q  = qs    [b];

    for (int t = 0; t < T; ++t) {
        qlp = aq * qlp + bq * q;
        const float x = fmaf(u, ufs, e + nx);

        // h = relu(bias + x*w_in + qlp*w_inq'); partial dot & ||h||^2
        float dot = 0.0f, hh = 0.0f;
#pragma unroll
        for (int k = 0; k < NPT_; ++k) {
            float hv = fmaf(x, win[k], fmaf(qlp, winq[k], bias[k]));
            hv = fmaxf(hv, 0.0f);
            h[k] = hv;
            dot  = fmaf(w[k], hv, dot);
            hh   = fmaf(hv,  hv, hh);
        }

        // wave32 butterfly reduction (ds_swizzle); two independent chains
#pragma unroll
        for (int m = 16; m >= 1; m >>= 1) {
            dot += __shfl_xor(dot, m, 32);
            hh  += __shfl_xor(hh,  m, 32);
        }

        const int buf = t & 1;
        if (lane == 0) partials[buf][wave] = make_float2(dot, hh);

        // --- split barrier: signal, fill the gap with independent work ---
        wg_barrier_signal();

        const int tn = (t + 1 < T) ? (t + 1) : t;     // clamped next index
        const int in = tn * BS_ + b;
        const float yn  = ys    [in];
        const float nxn = noises[in];
        const float qn  = qs    [in];
        if (t + PF_AHEAD < T) {                        // gfx1250 global_prefetch_b8
            const int ip = (t + PF_AHEAD) * BS_ + b;
            __builtin_prefetch(&ys    [ip], 0, 0);
            __builtin_prefetch(&noises[ip], 0, 0);
            __builtin_prefetch(&qs    [ip], 0, 0);
        }

        wg_barrier_wait();

        dot = 0.0f; hh = 0.0f;
#pragma unroll
        for (int wv = 0; wv < NWAVES; ++wv) {
            const float2 p = partials[buf][wv];
            dot += p.x; hh += p.y;
        }
        // (reuse of this buffer at t+2 is fenced: our read drains before we
        //  signal barrier t+1, and writers of t+2 first pass wait t+1)

        // ---- scalar chain (replicated, deterministic) ------------------
        const float u_new = dot;
        const float y_    = __builtin_isnan(y) ? u_new : y;
        ylp = ay * ylp + by * y_;
        const float e_new = ylp - u_new;
        elp = ae * elp + be * e_new;
        const float lr     = lr0 * lrm;
        const float lrelp  = lr * elp;
        const float lrspwd = lr * spwd;

        // w_out = (w_out + dw) - (lr*spwd)*w_out, dw = lr*elp*h
#pragma unroll
        for (int k = 0; k < NPT_; ++k) {
            const float wold = w[k];
            w[k] = fmaf(-lrspwd, wold, fmaf(lrelp, h[k], wold));
        }

        // ||dw||^2 = (lr*elp)^2 * sum h^2 ; norms = sqrt(fudge + ...)
        const float norm = sqrtf(fmaf(lrelp * lrelp, hh, 1e-4f));
        // argument is always small, negative, finite -> native v_exp_f32 path
        lrm *= __expf(-lrd * norm);

        u = u_new; e = e_new;

        if (tid == 0) out[t * BS_ + b] = os * u_new;

        y = yn; nx = nxn; q = qn;
    }
}

extern "C" void kernel_launch(void* const* d_in, const int* in_sizes, int n_in,
                              void* d_out, int out_size, void* d_ws, size_t ws_size,
                              hipStream_t stream) {
    // setup_inputs() order:
    //  0:n  1:noises  2:ys  3:qs  4:z_biases  5:w_in  6:w_inq
    //  7:log_learning_rate 8:log_learning_rate_decay 9:sigma_b 10:output_scale
    // 11:u_feedback_scale 12:sp_weight_decay 13:q_scale
    // 14:tauqlpf_m1 15:tauylpf_m1 16:tauelpf_m1
    (void)n_in; (void)d_ws; (void)ws_size; (void)out_size;

    const float* noises = (const float*)d_in[1];
    const float* ys     = (const float*)d_in[2];
    const float* qs     = (const float*)d_in[3];
    const float* zb     = (const float*)d_in[4];
    const float* wi     = (const float*)d_in[5];
    const float* wiq    = (const float*)d_in[6];

    const int T = in_sizes[1] / BS_;      // noises is [T, BS]

    elbo_scan_kernel<<<dim3(BS_), dim3(BLOCK_), 0, stream>>>(
        noises, ys, qs, zb, wi, wiq,
        (const float*)d_in[7],  (const float*)d_in[8],  (const float*)d_in[9],
        (const float*)d_in[10], (const float*)d_in[11], (const float*)d_in[12],
        (const float*)d_in[13], (const float*)d_in[14], (const float*)d_in[15],
        (const float*)d_in[16],
        (float*)d_out, T);
}